// AssocScan_15530601742938
// MI455X (gfx1250) — compile-verified
//
#include <hip/hip_runtime.h>

// Linear recurrence scan out[t] = g[t]*out[t-1] + x[t] over axis t (N=4096),
// independent per (b,d) channel. Memory-bound: 192 MB min HBM traffic
// (~8.2 us at 23.3 TB/s). 3-pass chunked scan; passes 1/3 use the CDNA5
// async global->LDS copy path (global_load_async_to_lds_b128 +
// s_wait_asynccnt) with double buffering. gates+inputs (128 MB) stay resident
// in the 192 MB L2 between pass 1 and pass 3; pass-3 output stores are
// non-temporal so they don't evict the input working set.

namespace {
constexpr int kB     = 4;
constexpr int kN     = 4096;
constexpr int kD     = 1024;
constexpr int kChunk = 32;               // timesteps per block
constexpr int kNCh   = kN / kChunk;      // 128 chunks along time
constexpr int kTT    = 4;                // timesteps staged per LDS tile
constexpr int kTiles = kChunk / kTT;     // 8 tiles per chunk
constexpr int kTPB   = 256;              // 8 waves; 4 channels (float4) / thread
constexpr int kAgg   = kB * kNCh * kD;   // elements per aggregate array (512K)
}  // namespace

// Native clang vector (required by __builtin_nontemporal_store).
typedef __attribute__((ext_vector_type(4))) float vfloat4;

// Per-lane 16B async copy global -> LDS (ASYNCcnt-tracked, no VGPR data).
__device__ __forceinline__ void async_copy_b128_to_lds(const float* gsrc,
                                                       unsigned lds_byte) {
  asm volatile("global_load_async_to_lds_b128 %0, %1, off"
               :
               : "v"(lds_byte), "v"(gsrc)
               : "memory");
}

template <int N>
__device__ __forceinline__ void wait_asynccnt_le() {
  asm volatile("s_wait_asynccnt %0" : : "n"(N) : "memory");
}

// kFinal=false: pass 1 -> per-chunk aggregates (A = prod g, X = local end state)
// kFinal=true : pass 3 -> seed h with carry, stream final outputs (NT stores)
template <bool kFinal>
__global__ __launch_bounds__(kTPB) void scan_chunks(
    const float* __restrict__ gates, const float* __restrict__ inputs,
    float* __restrict__ wsA, float* __restrict__ wsX,
    const float* __restrict__ wsC, float* __restrict__ out) {
  __shared__ float ldsG[2 * kTT * kD];   // 32 KB
  __shared__ float ldsX[2 * kTT * kD];   // 32 KB

  const int tid = threadIdx.x;
  const int d4  = tid << 2;                        // 4 consecutive channels
  const int b   = blockIdx.x / kNCh;
  const int c   = blockIdx.x % kNCh;
  const size_t base = ((size_t)b * kN + (size_t)c * kChunk) * kD + d4;
  const size_t agg  = (size_t)(b * kNCh + c) * kD + d4;

  const float* gbase = gates + base;
  const float* xbase = inputs + base;
  // Low 32 bits of a generic pointer to LDS == wave-relative LDS byte address.
  const unsigned lgBase = (unsigned)(size_t)&ldsG[d4];
  const unsigned lxBase = (unsigned)(size_t)&ldsX[d4];

  auto issue_tile = [&](int tile, int buf) {
    const float* gp = gbase + (size_t)(tile * kTT) * kD;
    const float* xp = xbase + (size_t)(tile * kTT) * kD;
    const unsigned lg = lgBase + (unsigned)(buf * kTT * kD) * 4u;
    const unsigned lx = lxBase + (unsigned)(buf * kTT * kD) * 4u;
#pragma unroll
    for (int tt = 0; tt < kTT; ++tt) {
      async_copy_b128_to_lds(gp + (size_t)tt * kD, lg + (unsigned)(tt * kD) * 4u);
      async_copy_b128_to_lds(xp + (size_t)tt * kD, lx + (unsigned)(tt * kD) * 4u);
    }
  };

  float4 A = make_float4(1.f, 1.f, 1.f, 1.f);
  float4 h;
  if constexpr (kFinal) {
    h = *(const float4*)&wsC[agg];                 // carry-in from pass 2
  } else {
    h = make_float4(0.f, 0.f, 0.f, 0.f);
  }

  issue_tile(0, 0);
  for (int tile = 0; tile < kTiles; ++tile) {
    const int buf = tile & 1;
    if (tile + 1 < kTiles) {
      issue_tile(tile + 1, buf ^ 1);
      // Async loads complete in order: <= 2*kTT outstanding means everything
      // older than the tile just issued (i.e. the current tile) is in LDS.
      wait_asynccnt_le<2 * kTT>();
    } else {
      wait_asynccnt_le<0>();
    }
    // Each lane reads back exactly the bytes it staged -> no barrier needed.
    const float* lg = &ldsG[buf * kTT * kD + d4];
    const float* lx = &ldsX[buf * kTT * kD + d4];
#pragma unroll
    for (int tt = 0; tt < kTT; ++tt) {
      const float4 g = *(const float4*)(lg + tt * kD);
      const float4 x = *(const float4*)(lx + tt * kD);
      h.x = fmaf(g.x, h.x, x.x);
      h.y = fmaf(g.y, h.y, x.y);
      h.z = fmaf(g.z, h.z, x.z);
      h.w = fmaf(g.w, h.w, x.w);
      if constexpr (kFinal) {
        // Non-temporal: output is write-once dead data; keep L2 for inputs.
        __builtin_nontemporal_store(
            *(const vfloat4*)&h,
            (vfloat4*)&out[base + (size_t)(tile * kTT + tt) * kD]);
      } else {
        A.x *= g.x; A.y *= g.y; A.z *= g.z; A.w *= g.w;
      }
    }
  }

  if constexpr (!kFinal) {
    *(float4*)&wsA[agg] = A;
    *(float4*)&wsX[agg] = h;
  }
}

// Pass 2: sequential scan over the 128 chunk aggregates per channel.
// carry[chunk c] = state entering chunk c (H_{-1} = 0; H_c = A_c*H_{c-1} + X_c).
__global__ __launch_bounds__(kTPB) void scan_carries(
    const float* __restrict__ wsA, const float* __restrict__ wsX,
    float* __restrict__ wsC) {
  const int b  = blockIdx.x;
  const int d4 = threadIdx.x << 2;
  float4 h = make_float4(0.f, 0.f, 0.f, 0.f);
  for (int c = 0; c < kNCh; ++c) {
    const size_t idx = (size_t)(b * kNCh + c) * kD + d4;
    *(float4*)&wsC[idx] = h;
    const float4 A = *(const float4*)&wsA[idx];
    const float4 X = *(const float4*)&wsX[idx];
    h.x = fmaf(A.x, h.x, X.x);
    h.y = fmaf(A.y, h.y, X.y);
    h.z = fmaf(A.z, h.z, X.z);
    h.w = fmaf(A.w, h.w, X.w);
  }
}

extern "C" void kernel_launch(void* const* d_in, const int* in_sizes, int n_in,
                              void* d_out, int out_size, void* d_ws, size_t ws_size,
                              hipStream_t stream) {
  (void)in_sizes; (void)n_in; (void)out_size; (void)ws_size;
  const float* gates  = (const float*)d_in[0];
  const float* inputs = (const float*)d_in[1];
  float* out = (float*)d_out;

  float* wsA = (float*)d_ws;       // [B][NCh][D] chunk gate products   (2 MB)
  float* wsX = wsA + (size_t)kAgg; // [B][NCh][D] chunk local end state (2 MB)
  float* wsC = wsX + (size_t)kAgg; // [B][NCh][D] carries into chunks   (2 MB)

  scan_chunks<false><<<kB * kNCh, kTPB, 0, stream>>>(gates, inputs, wsA, wsX,
                                                     nullptr, nullptr);
  scan_carries<<<kB, kTPB, 0, stream>>>(wsA, wsX, wsC);
  scan_chunks<true><<<kB * kNCh, kTPB, 0, stream>>>(gates, inputs, nullptr,
                                                    nullptr, wsC, out);
}